// GCNConv_58703613001790
// MI455X (gfx1250) — compile-verified
//
#include <hip/hip_runtime.h>
#include <hip/hip_bf16.h>

typedef __attribute__((ext_vector_type(2))) float v2f;
typedef __attribute__((ext_vector_type(8))) float v8f;

#define GCN_N   10000
#define GCN_E   320000
#define GCN_C   256          // in channels == out channels
#define BM_WORDS 3125000u    // ceil(N*N/32)

// ---------------- phase 0: zero the dedup bitmap -------------------------
__global__ void gcn_zero_words(unsigned int* __restrict__ p, unsigned int n) {
    unsigned int i = blockIdx.x * blockDim.x + threadIdx.x;
    unsigned int stride = gridDim.x * blockDim.x;
    for (; i < n; i += stride) p[i] = 0u;
}

// ---------------- phase 1: self-loop diagonal bits + deg=1 ---------------
// diagonal bit index = i*(N+1); consecutive diagonals are 10001 bits apart
// (> 32), so no two diagonal bits share a word -> plain store into the
// freshly zeroed bitmap is race-free.
__global__ void gcn_init_diag(unsigned int* __restrict__ bitmap,
                              unsigned int* __restrict__ deg) {
    int i = blockIdx.x * blockDim.x + threadIdx.x;
    if (i < GCN_N) {
        unsigned int idx = (unsigned int)i * (unsigned int)(GCN_N + 1);
        bitmap[idx >> 5] = (1u << (idx & 31u));
        deg[i] = 1u;
    }
}

// ---------------- phase 2: dedup edges, accumulate degrees ---------------
__global__ void gcn_edge_pass(const long long* __restrict__ ei,
                              unsigned int* __restrict__ bitmap,
                              unsigned int* __restrict__ deg,
                              unsigned char* __restrict__ uniq) {
    int e = blockIdx.x * blockDim.x + threadIdx.x;
    if (e >= GCN_E) return;
    int r = (int)ei[e];
    int c = (int)ei[GCN_E + e];
    unsigned int idx  = (unsigned int)r * (unsigned int)GCN_N + (unsigned int)c;
    unsigned int mask = 1u << (idx & 31u);
    unsigned int old  = atomicOr(&bitmap[idx >> 5], mask);
    if ((old & mask) == 0u) {          // first writer of this adjacency cell
        uniq[e] = 1;
        atomicAdd(&deg[r], 1u);
    } else {
        uniq[e] = 0;                   // duplicate edge (or self-loop dup)
    }
}

// ---------------- phase 3: d^{-1/2} -------------------------------------
__global__ void gcn_deg_rsqrt(const unsigned int* __restrict__ deg,
                              float* __restrict__ dinv) {
    int i = blockIdx.x * blockDim.x + threadIdx.x;
    if (i < GCN_N) dinv[i] = rsqrtf((float)deg[i]);
}

// ---------------- phase 4: Y = dinv[i]^2 * X[i]  (self-loop term) --------
// float4 over N*C/4 elements; also serves as Y's initialization.
__global__ void gcn_init_y(const float* __restrict__ x,
                           const float* __restrict__ dinv,
                           float* __restrict__ y) {
    int i = blockIdx.x * blockDim.x + threadIdx.x;   // over N*64 float4s
    if (i >= GCN_N * (GCN_C / 4)) return;
    int node = i >> 6;
    float s = dinv[node];
    s = s * s;
    float4 xv = ((const float4*)x)[i];
    float4 yv;
    yv.x = s * xv.x; yv.y = s * xv.y; yv.z = s * xv.z; yv.w = s * xv.w;
    ((float4*)y)[i] = yv;
}

// ---------------- phase 5: scatter  Y[r] += d[r]*d[c]*X[c] ---------------
// 4 edges per 256-thread block; 64 lanes per edge, float4 per lane.
__global__ void gcn_scatter(const long long* __restrict__ ei,
                            const float* __restrict__ x,
                            const float* __restrict__ dinv,
                            const unsigned char* __restrict__ uniq,
                            float* __restrict__ y) {
    int e = blockIdx.x * 4 + (threadIdx.x >> 6);
    if (e >= GCN_E) return;
    if (!uniq[e]) return;
    int r = (int)ei[e];
    int c = (int)ei[GCN_E + e];
    float s = dinv[r] * dinv[c];
    int k = (threadIdx.x & 63) * 4;
    float4 xv = *(const float4*)(x + c * GCN_C + k);
    float* yp = y + r * GCN_C + k;
    atomicAdd(yp + 0, s * xv.x);
    atomicAdd(yp + 1, s * xv.y);
    atomicAdd(yp + 2, s * xv.z);
    atomicAdd(yp + 3, s * xv.w);
}

// ---------------- phase 6: out = Y @ W via V_WMMA_F32_16X16X4_F32 --------
// Block = 128 threads (4 waves). Block b owns rows [16b,16b+16); wave w owns
// columns [64w, 64w+64) as four 16x16 accumulator tiles, reusing the A
// fragment across the 4 tiles. K loops 0..255 step 4.
//
// ISA 7.12.2 layouts (wave32):
//   A 16x4 f32 : lane<16 -> M=lane,    v0=K0, v1=K1
//                lane>=16 -> M=lane-16, v0=K2, v1=K3
//   B 4x16 f32 : mirrored (lane<16 -> N=lane, v0=K0,v1=K1; lane>=16 K2/K3)
//   C/D 16x16  : 8 VGPRs; VGPR v holds M = v + 8*(lane>=16), N = lane&15
__global__ void gcn_gemm_wmma(const float* __restrict__ y,
                              const float* __restrict__ w,
                              float* __restrict__ out) {
    const int row0 = blockIdx.x * 16;
    const int wave = threadIdx.x >> 5;
    const int lane = threadIdx.x & 31;
    const int half = lane >> 4;       // 0: K0/K1 (or M 0..7), 1: K2/K3 (M 8..15)
    const int lrow = lane & 15;       // M for A, N for B/C/D
    const int n0   = wave * 64;

    v8f c0 = {}, c1 = {}, c2 = {}, c3 = {};
    const float* arow = y + (row0 + lrow) * GCN_C;

    for (int k0 = 0; k0 < GCN_C; k0 += 4) {
        const float2 a2 = *(const float2*)(arow + k0 + 2 * half);
        v2f a; a.x = a2.x; a.y = a2.y;
        const float* wp = w + (k0 + 2 * half) * GCN_C;  // row k0+2h of W
#pragma unroll
        for (int t = 0; t < 4; ++t) {
            int ncol = n0 + t * 16 + lrow;
            v2f b; b.x = wp[ncol]; b.y = wp[GCN_C + ncol];
            v8f& c = (t == 0 ? c0 : t == 1 ? c1 : t == 2 ? c2 : c3);
            c = __builtin_amdgcn_wmma_f32_16x16x4_f32(
                    /*neg_a=*/false, a, /*neg_b=*/false, b,
                    /*c_mod=*/(short)0, c, /*reuse_a=*/false, /*reuse_b=*/false);
        }
    }

#pragma unroll
    for (int t = 0; t < 4; ++t) {
        const v8f& c = (t == 0 ? c0 : t == 1 ? c1 : t == 2 ? c2 : c3);
#pragma unroll
        for (int v = 0; v < 8; ++v) {
            out[(row0 + v + 8 * half) * GCN_C + n0 + t * 16 + lrow] = c[v];
        }
    }
}

extern "C" void kernel_launch(void* const* d_in, const int* in_sizes, int n_in,
                              void* d_out, int out_size, void* d_ws, size_t ws_size,
                              hipStream_t stream) {
    const float*     x  = (const float*)d_in[0];        // (N, 256) f32
    const float*     w  = (const float*)d_in[1];        // (256, 256) f32
    const long long* ei = (const long long*)d_in[2];    // (2, E) i64
    float* out = (float*)d_out;                         // (N, 256) f32

    char* ws = (char*)d_ws;
    // layout: [bitmap 12.5MB][deg 40KB][dinv 40KB][uniq 320KB][pad][Y 10.24MB]
    unsigned int*  bitmap = (unsigned int*)(ws);
    unsigned int*  deg    = (unsigned int*)(ws + 12500000);
    float*         dinv   = (float*)       (ws + 12540000);
    unsigned char* uniq   = (unsigned char*)(ws + 12580000);
    float*         ybuf   = (float*)       (ws + 12900096);   // 16B aligned

    gcn_zero_words<<<4096, 256, 0, stream>>>(bitmap, BM_WORDS);
    gcn_init_diag<<<(GCN_N + 255) / 256, 256, 0, stream>>>(bitmap, deg);
    gcn_edge_pass<<<(GCN_E + 255) / 256, 256, 0, stream>>>(ei, bitmap, deg, uniq);
    gcn_deg_rsqrt<<<(GCN_N + 255) / 256, 256, 0, stream>>>(deg, dinv);
    gcn_init_y<<<(GCN_N * (GCN_C / 4) + 255) / 256, 256, 0, stream>>>(x, dinv, ybuf);
    gcn_scatter<<<GCN_E / 4, 256, 0, stream>>>(ei, x, dinv, uniq, ybuf);
    gcn_gemm_wmma<<<GCN_N / 16, 128, 0, stream>>>(ybuf, w, out);
}